// ODEFunc_16071767622283
// MI455X (gfx1250) — compile-verified
//
#include <hip/hip_runtime.h>
#include <stdint.h>

// Problem constants (match reference)
#define N_NODES  100000
#define N_EDGES  1600000
#define HIDDEN   64
#define CHUNK    1024            // edges staged per TDM chunk (4KB per array)
#define BLOCK_T  256             // 8 wave32 waves
#define NWAVES   (BLOCK_T / 32)
#define PER_WAVE (CHUNK / NWAVES) // 128 edges per wave per chunk

typedef __attribute__((ext_vector_type(4))) unsigned int v4u;
typedef __attribute__((ext_vector_type(8))) int          v8i;
typedef __attribute__((ext_vector_type(4))) int          v4i;

// ---------------------------------------------------------------------------
// Issue one TDM 1-D copy: n_valid 4-byte elements available at gptr, tile of
// CHUNK elements into LDS at lds_addr. Reads past tensor_dim0 return zero,
// which handles the tail chunk. D# bit packing per CDNA5 ISA
// ---------------------------------------------------------------------------
__device__ __forceinline__ void tdm_load_1d(unsigned lds_addr, const void* gptr,
                                            unsigned n_valid) {
    unsigned long long ga = (unsigned long long)(uintptr_t)gptr;

    v4u g0;
    g0[0] = 1u;                                     // count=1, is_restore=0, gather=0
    g0[1] = lds_addr;                               // lds_addr [63:32]
    g0[2] = (unsigned)(ga & 0xFFFFFFFFull);         // global_addr lo32
    g0[3] = ((unsigned)(ga >> 32) & 0x01FFFFFFu)    // global_addr [56:32]
            | 0x80000000u;                          // type=2 ("image")

    v8i g1;
    g1[0] = (int)0x00020000u;                       // wg_mask=0, data_size=2 (4B)
    g1[1] = (int)((n_valid & 0xFFFFu) << 16);       // tensor_dim0 lo16
    g1[2] = (int)(((n_valid >> 16) & 0xFFFFu)       // tensor_dim0 hi16
                  | (1u << 16));                    // tensor_dim1 = 1 (lo16)
    g1[3] = (int)((unsigned)CHUNK << 16);           // tensor_dim1 hi=0, tile_dim0=CHUNK
    g1[4] = 1;                                      // tile_dim1 = 1, tile_dim2 = 0
    g1[5] = (int)n_valid;                           // tensor_dim0_stride lo32 (1 line: unused)
    g1[6] = 0;                                      // stride0 hi16, stride1 lo16
    g1[7] = 0;                                      // stride1 hi32

    v4i gz4  = {0, 0, 0, 0};                        // groups 2/3: dims >2 unused
    v8i gz8  = {0, 0, 0, 0, 0, 0, 0, 0};

    __builtin_amdgcn_tensor_load_to_lds(g0, g1, gz4, gz4, gz8, 0);
}

// ---------------------------------------------------------------------------
// Kernel: COO scatter-add SpMM.  (Placed first so the disasm snippet shows it.)
//  - Each block grid-strides over CHUNK-edge chunks.
//  - Wave 0 drives TDM double buffering of (rows, cols, vals) into LDS.
//  - One wave processes one edge per step: 32 lanes x float2 = 64 hidden.
// ---------------------------------------------------------------------------
__global__ __launch_bounds__(BLOCK_T) void spmm_scatter_kernel(
    const float* __restrict__ x,
    const int*   __restrict__ A_rows,
    const int*   __restrict__ A_cols,
    const float* __restrict__ A_vals,
    float*       __restrict__ out)
{
    __shared__ int   s_rows[2][CHUNK];
    __shared__ int   s_cols[2][CHUNK];
    __shared__ float s_vals[2][CHUNK];

    const int nchunks = (N_EDGES + CHUNK - 1) / CHUNK;
    const int wave = threadIdx.x >> 5;
    const int lane = threadIdx.x & 31;

    const int first = blockIdx.x;

    // Prologue: wave 0 preloads this block's first chunk into buffer 0.
    if (first < nchunks && threadIdx.x < 32) {
        unsigned nrem = (unsigned)(N_EDGES - first * CHUNK);
        tdm_load_1d((unsigned)(uintptr_t)&s_rows[0][0], A_rows + (size_t)first * CHUNK, nrem);
        tdm_load_1d((unsigned)(uintptr_t)&s_cols[0][0], A_cols + (size_t)first * CHUNK, nrem);
        tdm_load_1d((unsigned)(uintptr_t)&s_vals[0][0], A_vals + (size_t)first * CHUNK, nrem);
    }

    const float2* __restrict__ xp = (const float2*)x;
    int buf = 0;

    for (int c = first; c < nchunks; c += (int)gridDim.x) {
        // Wave 0 waits for the TDM fills of buf; others have TENSORcnt==0 (no-op).
        __builtin_amdgcn_s_wait_tensorcnt(0);
        // Barrier: (a) publishes buf to all waves, (b) proves all waves finished
        // reading buf^1 in the previous iteration, so it is safe to refill it.
        __syncthreads();

        int nxt = c + (int)gridDim.x;
        if (nxt < nchunks && threadIdx.x < 32) {
            unsigned nrem = (unsigned)(N_EDGES - nxt * CHUNK);
            int nb = buf ^ 1;
            tdm_load_1d((unsigned)(uintptr_t)&s_rows[nb][0], A_rows + (size_t)nxt * CHUNK, nrem);
            tdm_load_1d((unsigned)(uintptr_t)&s_cols[nb][0], A_cols + (size_t)nxt * CHUNK, nrem);
            tdm_load_1d((unsigned)(uintptr_t)&s_vals[nb][0], A_vals + (size_t)nxt * CHUNK, nrem);
        }

        // Consume buf: wave w owns edges [w*PER_WAVE, w*PER_WAVE + ecnt).
        int base  = c * CHUNK;
        int valid = N_EDGES - base; if (valid > CHUNK) valid = CHUNK;
        int e0    = wave * PER_WAVE;
        int ecnt  = valid - e0; if (ecnt > PER_WAVE) ecnt = PER_WAVE;

#pragma unroll 4
        for (int k = 0; k < ecnt; ++k) {
            int   e   = e0 + k;
            int   row = s_rows[buf][e];      // LDS broadcast (all lanes same addr)
            int   col = s_cols[buf][e];
            float v   = s_vals[buf][e];
            float2 xv = xp[(size_t)col * (HIDDEN / 2) + lane];  // global_load_b64 gather
            float* dst = out + (size_t)row * HIDDEN + lane * 2;
            atomicAdd(dst + 0, v * xv.x);    // global_atomic_add_f32 (no return)
            atomicAdd(dst + 1, v * xv.y);
        }
        buf ^= 1;
    }
}

// ---------------------------------------------------------------------------
// Zero the accumulator (d_out is poisoned by the harness)
// ---------------------------------------------------------------------------
__global__ void zero_kernel(float4* __restrict__ p, int n4) {
    int i = blockIdx.x * blockDim.x + threadIdx.x;
    if (i < n4) p[i] = make_float4(0.f, 0.f, 0.f, 0.f);
}

// ---------------------------------------------------------------------------
// In-place ReLU
// ---------------------------------------------------------------------------
__global__ void relu_kernel(float4* __restrict__ p, int n4) {
    int i = blockIdx.x * blockDim.x + threadIdx.x;
    if (i < n4) {
        float4 v = p[i];
        v.x = fmaxf(v.x, 0.f);
        v.y = fmaxf(v.y, 0.f);
        v.z = fmaxf(v.z, 0.f);
        v.w = fmaxf(v.w, 0.f);
        p[i] = v;
    }
}

extern "C" void kernel_launch(void* const* d_in, const int* in_sizes, int n_in,
                              void* d_out, int out_size, void* d_ws, size_t ws_size,
                              hipStream_t stream) {
    // inputs: 0=t (unused scalar), 1=x, 2=A_rows, 3=A_cols, 4=A_vals
    const float* x      = (const float*)d_in[1];
    const int*   A_rows = (const int*)  d_in[2];
    const int*   A_cols = (const int*)  d_in[3];
    const float* A_vals = (const float*)d_in[4];
    float*       out    = (float*)d_out;

    const int n4 = (N_NODES * HIDDEN) / 4;   // 1.6M float4
    zero_kernel<<<(n4 + 255) / 256, 256, 0, stream>>>((float4*)out, n4);

    const int nchunks = (N_EDGES + CHUNK - 1) / CHUNK;
    int grid = 512;                          // ~3 chunks per block -> pipeline active
    if (grid > nchunks) grid = nchunks;
    spmm_scatter_kernel<<<grid, BLOCK_T, 0, stream>>>(x, A_rows, A_cols, A_vals, out);

    relu_kernel<<<(n4 + 255) / 256, 256, 0, stream>>>((float4*)out, n4);
}